// DimensionPruning_4655744549089
// MI455X (gfx1250) — compile-verified
//
#include <hip/hip_runtime.h>
#include <hip/hip_bf16.h>
#include <limits.h>

#define N_OBJ 100000
#define N_DIM 512
#define HSTRIDE 100096            // buckets 0..100000, padded to multiple of 128
#define NCHUNK (HSTRIDE / 128)    // 782 chunks of 128 buckets per dim

typedef __attribute__((ext_vector_type(2))) float v2f;
typedef __attribute__((ext_vector_type(8))) float v8f;

// threshold t_k = fl(fl(k/n) * alpha), matching jnp float32 arithmetic
__device__ __forceinline__ float thr(int k) {
    return ((float)k / 100000.0f) * 0.05f;
}

__device__ __forceinline__ float bcast_lane(float v, int lane) {
    return __int_as_float(__builtin_amdgcn_readlane(__float_as_int(v), lane));
}

// ---------------- kernel 1: zero the histogram workspace ----------------
__global__ __launch_bounds__(256) void k_zero(uint4* H4, long n4) {
    long stride = (long)gridDim.x * blockDim.x;
    for (long i = (long)blockIdx.x * blockDim.x + threadIdx.x; i < n4; i += stride)
        H4[i] = make_uint4(0u, 0u, 0u, 0u);
}

// ---------------- kernel 2: p-value -> bucket -> atomic histogram ----------------
__global__ __launch_bounds__(256) void k_hist(const float* __restrict__ q_mu,
                                              const float* __restrict__ q_var,
                                              unsigned int* __restrict__ H) {
    long gid = (long)blockIdx.x * blockDim.x + threadIdx.x;
    if (gid >= (long)N_OBJ * N_DIM) return;
    int dim = (int)(gid & (N_DIM - 1));           // row-major [obj][dim]
    float mu = q_mu[gid];
    float sg = q_var[gid];
    float p = normcdff(-mu / sg);                 // Phi(0; mu, sigma)
    if (!(p <= 0.05f)) return;                    // can never be rejected (also NaN guard)
    // bucket b(p) = min{k >= 0 : p <= thr(k)}  (thr(0)=0 catches p==0 underflow)
    int k0 = (int)ceilf(p * 2.0e6f);              // guess: p * n / alpha
    if (k0 < 0) k0 = 0;
    if (k0 > N_OBJ) k0 = N_OBJ;
    while (k0 > 0 && p <= thr(k0 - 1)) --k0;
    while (p > thr(k0)) ++k0;                     // terminates: p <= thr(N_OBJ)
    atomicAdd(&H[(size_t)dim * HSTRIDE + k0], 1u);
}

// ---------------- kernel 3: WMMA-accelerated prefix scan + BH condition ----------------
// One wave per dim, 128 buckets/iteration (uint4 per lane). The 32 per-lane group
// sums are scanned with V_WMMA_F32_16X16X4_F32: pre-shuffle src(L)=L<16?2L:2L-31
// makes row m sum positions (2m,2m+1); with B = all-ones, D row m = pairsum(m).
// Per the ISA C/D layout, lanes 0-15 receive pairsums 0-7 and lanes 16-31 receive
// 8-15 — exactly the pair prefixes each lane needs; readlane supplies the
// half totals. All values are small integers, exact in f32.
// Software-pipelined depth 2 + WGP-scope prefetch 16 chunks ahead.
__global__ __launch_bounds__(32) void k_scan(const unsigned int* __restrict__ H,
                                             float* __restrict__ out) {
    int d = blockIdx.x;
    const unsigned int* h = H + (size_t)d * HSTRIDE;
    int lane = threadIdx.x;

    int srcL = (lane < 16) ? (2 * lane) : (2 * lane - 31);
    int lim  = (lane < 16) ? (lane >> 1) : ((lane >> 1) - 8);  // my pair index within my half
    bool odd = (lane & 1) != 0;
    bool hi  = lane >= 16;

    v2f bmat; bmat.x = 1.0f; bmat.y = 1.0f;     // B = all ones (layout independent)

    float offset = 0.0f;                         // running prefix P before this chunk
    int cnt = 0;
    int kmin = INT_MAX;

    // depth-2 software pipeline: two b128 loads in flight
    uint4 buf0 = *(const uint4*)(h + 0 * 128 + 4 * lane);
    uint4 buf1 = *(const uint4*)(h + 1 * 128 + 4 * lane);

    for (int c = 0; c < NCHUNK; ++c) {
        int cn = c + 2 < NCHUNK ? c + 2 : NCHUNK - 1;
        uint4 nxt = *(const uint4*)(h + cn * 128 + 4 * lane);
        if (c + 16 < NCHUNK)
            __builtin_prefetch(h + (c + 16) * 128 + 4 * lane, 0, 3); // global_prefetch_b8 (WGP)

        int base = c * 128;
        float w0 = (float)buf0.x, w1 = (float)buf0.y, w2 = (float)buf0.z, w3 = (float)buf0.w;
        float gsum = w0 + w1 + w2 + w3;          // this lane's 4-bucket group sum

        // feed permutation for the WMMA pair structure
        float fv = __shfl(gsum, srcL, 32);
        v2f a; a.x = fv; a.y = 0.0f;
        v8f cacc = {};
        v8f dacc = __builtin_amdgcn_wmma_f32_16x16x4_f32(
            false, a, false, bmat, (short)0, cacc, false, false);

        // dacc[r] = pairsum(r) (lanes 0-15) or pairsum(r+8) (lanes 16-31)
        float s = 0.0f, hsum = 0.0f, ownp = 0.0f;
#pragma unroll
        for (int r = 0; r < 8; ++r) {
            float pr = dacc[r];
            hsum += pr;
            if (r < lim)  s += pr;
            if (r == lim) ownp = pr;
        }
        float T0 = bcast_lane(hsum, 0);          // total of pairs 0-7  (positions 0-15)
        float T1 = bcast_lane(hsum, 16);         // total of pairs 8-15 (positions 16-31)

        float excl = s + (odd ? (ownp - gsum) : 0.0f) + (hi ? T0 : 0.0f);
        float P0 = offset + excl;                // prefix before this lane's 4 buckets

        float P;  int k;
        k = base + 4 * lane;
        P = P0 + w0; if (k >= 1 && k <= N_OBJ && P >= (float)k) { ++cnt; kmin = min(kmin, k); }
        ++k;
        P += w1;     if (k <= N_OBJ && P >= (float)k)           { ++cnt; kmin = min(kmin, k); }
        ++k;
        P += w2;     if (k <= N_OBJ && P >= (float)k)           { ++cnt; kmin = min(kmin, k); }
        ++k;
        P += w3;     if (k <= N_OBJ && P >= (float)k)           { ++cnt; kmin = min(kmin, k); }

        offset += T0 + T1;                       // chunk total
        buf0 = buf1;
        buf1 = nxt;
    }

    // wave reduction of per-lane count / first-k
#pragma unroll
    for (int o = 16; o > 0; o >>= 1) {
        cnt  += __shfl_xor(cnt, o, 32);
        kmin  = min(kmin, __shfl_xor(kmin, o, 32));
    }
    if (lane == 0)
        out[d] = (kmin == INT_MAX) ? 0.0f : (float)(cnt + kmin - 1);
}

// ---------------- launcher ----------------
extern "C" void kernel_launch(void* const* d_in, const int* in_sizes, int n_in,
                              void* d_out, int out_size, void* d_ws, size_t ws_size,
                              hipStream_t stream) {
    const float* q_mu  = (const float*)d_in[0];
    const float* q_var = (const float*)d_in[1];
    float* out = (float*)d_out;
    unsigned int* H = (unsigned int*)d_ws;       // 512 * 100096 u32 ~ 195.5 MiB

    long n4 = ((long)N_DIM * HSTRIDE) / 4;       // uint4 count (HSTRIDE % 4 == 0)
    k_zero<<<8192, 256, 0, stream>>>((uint4*)H, n4);

    long total = (long)N_OBJ * N_DIM;
    int blocks = (int)((total + 255) / 256);
    k_hist<<<blocks, 256, 0, stream>>>(q_mu, q_var, H);

    k_scan<<<N_DIM, 32, 0, stream>>>(H, out);
    (void)in_sizes; (void)n_in; (void)out_size; (void)ws_size;
}